// DistGATModel_39470749450536
// MI455X (gfx1250) — compile-verified
//
#include <hip/hip_runtime.h>

// ---------------------------------------------------------------------------
// fp32 WMMA fragment types (gfx1250, wave32)
//   A: 16x4  f32 -> 2 VGPRs/lane (lane m=lane&15, k = 2*(lane>>4) + {0,1})
//   B: 4x16  f32 -> 2 VGPRs/lane (lane n=lane&15, k = 2*(lane>>4) + {0,1})
//   C/D: 16x16 f32 -> 8 VGPRs/lane (row = v + 8*(lane>>4), col = lane&15)
// ---------------------------------------------------------------------------
typedef __attribute__((ext_vector_type(2))) float v2f;
typedef __attribute__((ext_vector_type(8))) float v8f;

__device__ __forceinline__ float lrelu(float x) { return x > 0.0f ? x : 0.2f * x; }

// ---------------------------------------------------------------------------
// GEMM: C[M,N] = A[M,K] * B[K,N], fp32, v_wmma_f32_16x16x4_f32.
// One wave computes a (16*MT) x (16*NT) block: per k-step, NT B-fragments are
// loaded once and reused by MT A-fragments -> MT*NT WMMAs per iteration.
// Single base pointers + offsets (not incremented pointer arrays) so the
// backend emits global_load (LOADcnt only), not flat_load (LOADcnt+DScnt).
// GUARD=false: N must be a multiple of 16*NT (no branches in the hot loop).
// GUARD=true : column guards via clamped address + value select (no saveexec
//              in the k-loop; EXEC stays all-ones around WMMA).
// M must be a multiple of 16*MT.
// ---------------------------------------------------------------------------
template <int MT, int NT, bool GUARD>
__launch_bounds__(256)
__global__ void gemm_wmma_f32(const float* __restrict__ A,
                              const float* __restrict__ B,
                              float* __restrict__ C,
                              int M, int K, int N) {
    const int lane = threadIdx.x & 31;
    const int wave = threadIdx.x >> 5;
    const int NG = (N + 16 * NT - 1) / (16 * NT);
    const int mtiles = M / (16 * MT);
    const long flat = (long)blockIdx.x * (blockDim.x >> 5) + wave;
    if (flat >= (long)mtiles * NG) return;          // wave-uniform
    const int mt = (int)(flat / NG);
    const int ng = (int)(flat % NG);

    const int row0 = mt * 16 * MT;
    const int col0 = ng * 16 * NT;
    const int m    = lane & 15;                      // M index (A), N index (B/C)
    const int kb   = (lane >> 4) * 2;                // K sub-offset 0 or 2

    v8f acc[MT][NT];
#pragma unroll
    for (int r = 0; r < MT; ++r)
#pragma unroll
        for (int t = 0; t < NT; ++t) acc[r][t] = (v8f){0, 0, 0, 0, 0, 0, 0, 0};

    // Single base pointers; all loads indexed off these (=> global_load).
    const float* __restrict__ abase = A + (size_t)(row0 + m) * K + kb;
    const float* __restrict__ bbase = B + (size_t)kb * N;

    int cidx[NT];       // clamped column index (GUARD) or exact column
    bool cok[NT];
#pragma unroll
    for (int t = 0; t < NT; ++t) {
        const int c = col0 + t * 16 + m;
        cok[t]  = GUARD ? (c < N) : true;
        cidx[t] = GUARD ? (c < N ? c : N - 1) : c;
    }

    for (int k0 = 0; k0 < K; k0 += 4) {
        v2f b[NT];
#pragma unroll
        for (int t = 0; t < NT; ++t) {
            const size_t bo = (size_t)k0 * N + cidx[t];
            float bx = bbase[bo];
            float by = bbase[bo + N];
            b[t].x = GUARD ? (cok[t] ? bx : 0.0f) : bx;
            b[t].y = GUARD ? (cok[t] ? by : 0.0f) : by;
        }
#pragma unroll
        for (int r = 0; r < MT; ++r) {
            const size_t ao = (size_t)r * 16 * K + k0;
            v2f a;
            a.x = abase[ao + 0];
            a.y = abase[ao + 1];
#pragma unroll
            for (int t = 0; t < NT; ++t)
                acc[r][t] = __builtin_amdgcn_wmma_f32_16x16x4_f32(
                    false, a, false, b[t], (short)0, acc[r][t], false, false);
        }
    }

    const int rhalf = (lane >> 4) * 8;
#pragma unroll
    for (int r = 0; r < MT; ++r) {
        const int rbase = row0 + r * 16 + rhalf;
#pragma unroll
        for (int t = 0; t < NT; ++t) {
            const int c = col0 + t * 16 + m;
            if (!GUARD || c < N) {
#pragma unroll
                for (int v = 0; v < 8; ++v)
                    C[(size_t)(rbase + v) * N + c] = acc[r][t][v];
            }
        }
    }
}

// ---------------------------------------------------------------------------
// Per-(node, head) attention logits: e[i,h] = sum_d h[i,h,d] * a[h,d]
// ---------------------------------------------------------------------------
__global__ void attn_logits(const float* __restrict__ h,
                            const float* __restrict__ avec,   // H*D
                            float* __restrict__ e,            // N*H
                            int N, int H, int D) {
    const int i = blockIdx.x * blockDim.x + threadIdx.x;
    if (i >= N * H) return;
    const int node = i / H;
    const int hh   = i % H;
    const float* __restrict__ row = h + (size_t)node * H * D + (size_t)hh * D;
    const float* __restrict__ av  = avec + (size_t)hh * D;
    float s = 0.0f;
    for (int d = 0; d < D; ++d) s += row[d] * av[d];
    e[i] = s;
}

// ---------------------------------------------------------------------------
// CSR row offsets from a SORTED dst array.
// ---------------------------------------------------------------------------
__global__ void build_offsets(const int* __restrict__ dst, int E, int n_dst,
                              int* __restrict__ offs) {
    const int e = blockIdx.x * blockDim.x + threadIdx.x;
    if (e >= E) return;
    const int d    = dst[e];
    const int prev = (e == 0) ? -1 : dst[e - 1];
    for (int r = prev + 1; r <= d; ++r) offs[r] = e;
    if (e == E - 1)
        for (int r = d + 1; r <= n_dst; ++r) offs[r] = E;
}

// ---------------------------------------------------------------------------
// Softmax-weighted neighbor aggregation. One wave (32 lanes) per dst node.
// Lanes stride the H*D feature dim (coalesced); edge loop is serial
// (avg degree ~10), two passes: (1) per-head max, (2) exp/sum/accumulate.
// Deterministic, no atomics (edges of a dst node are contiguous via CSR).
// ---------------------------------------------------------------------------
template <int H, int D>
__global__ void gat_aggregate(const float* __restrict__ h,     // N_src x H*D
                              const float* __restrict__ el,    // N_src x H
                              const float* __restrict__ er,    // n_dst x H
                              const int*  __restrict__ src,
                              const int*  __restrict__ offs,   // n_dst+1
                              float* __restrict__ out,         // n_dst x H*D
                              int n_dst, int do_relu) {
    constexpr int HD = H * D;
    constexpr int JN = (HD + 31) / 32;
    const int d    = blockIdx.x;
    const int lane = threadIdx.x;
    if (d >= n_dst) return;

    const int e0 = offs[d];
    const int e1 = offs[d + 1];

    if (e0 == e1) {   // isolated node -> zeros
#pragma unroll
        for (int j = 0; j < JN; ++j) {
            const int f = lane + j * 32;
            if (f < HD) out[(size_t)d * HD + f] = 0.0f;
        }
        return;
    }

    float erd[H];
#pragma unroll
    for (int hh = 0; hh < H; ++hh) erd[hh] = er[(size_t)d * H + hh];

    // pass 1: per-head running max (wave-uniform redundant compute)
    float mx[H];
#pragma unroll
    for (int hh = 0; hh < H; ++hh) mx[hh] = -1e30f;
    for (int e = e0; e < e1; ++e) {
        const int s = src[e];
        __builtin_prefetch(h + (size_t)s * HD, 0, 1);   // warm L2 for pass 2
#pragma unroll
        for (int hh = 0; hh < H; ++hh) {
            const float sc = lrelu(el[(size_t)s * H + hh] + erd[hh]);
            mx[hh] = fmaxf(mx[hh], sc);
        }
    }

    // pass 2: exp / sum / weighted feature accumulation
    float ssum[H];
#pragma unroll
    for (int hh = 0; hh < H; ++hh) ssum[hh] = 0.0f;
    float acc[JN];
#pragma unroll
    for (int j = 0; j < JN; ++j) acc[j] = 0.0f;

    for (int e = e0; e < e1; ++e) {
        const int s = src[e];
        float a[H];
#pragma unroll
        for (int hh = 0; hh < H; ++hh) {
            const float sc = lrelu(el[(size_t)s * H + hh] + erd[hh]);
            a[hh] = __expf(sc - mx[hh]);
            ssum[hh] += a[hh];
        }
        const float* __restrict__ hrow = h + (size_t)s * HD;
#pragma unroll
        for (int j = 0; j < JN; ++j) {
            const int f = lane + j * 32;
            if (f < HD) acc[j] += a[f / D] * hrow[f];
        }
    }

#pragma unroll
    for (int j = 0; j < JN; ++j) {
        const int f = lane + j * 32;
        if (f < HD) {
            float v = acc[j] / ssum[f / D];
            if (do_relu) v = fmaxf(v, 0.0f);
            out[(size_t)d * HD + f] = v;
        }
    }
}

// ---------------------------------------------------------------------------
// Host orchestration
// ---------------------------------------------------------------------------
template <int MT, int NT, bool GUARD>
static inline void launch_gemm(const float* A, const float* B, float* C,
                               int M, int K, int N, hipStream_t s) {
    const int threads = 256;                         // 8 waves
    const int NG = (N + 16 * NT - 1) / (16 * NT);
    const long waves = (long)(M / (16 * MT)) * NG;
    const int blocks = (int)((waves + 7) / 8);
    hipLaunchKernelGGL((gemm_wmma_f32<MT, NT, GUARD>), dim3(blocks), dim3(threads),
                       0, s, A, B, C, M, K, N);
}

extern "C" void kernel_launch(void* const* d_in, const int* in_sizes, int n_in,
                              void* d_out, int out_size, void* d_ws, size_t ws_size,
                              hipStream_t stream) {
    const float* x   = (const float*)d_in[0];
    const int* src0  = (const int*)d_in[1];
    const int* dst0  = (const int*)d_in[2];
    const int* src1  = (const int*)d_in[3];
    const int* dst1  = (const int*)d_in[4];
    const int* src2  = (const int*)d_in[5];
    const int* dst2  = (const int*)d_in[6];
    const float* W0  = (const float*)d_in[7];
    const float* al0 = (const float*)d_in[8];
    const float* ar0 = (const float*)d_in[9];
    const float* W1  = (const float*)d_in[10];
    const float* al1 = (const float*)d_in[11];
    const float* ar1 = (const float*)d_in[12];
    const float* W2  = (const float*)d_in[13];
    const float* al2 = (const float*)d_in[14];
    const float* ar2 = (const float*)d_in[15];
    float* outp = (float*)d_out;
    (void)in_sizes; (void)n_in; (void)out_size; (void)ws_size;

    const int N0 = 200000, N1 = 100000, N2 = 50000, N3 = 25000;
    const int E0 = 1000000, E1 = 500000, E2 = 250000;

    char* ws = (char*)d_ws;
    size_t off = 0;
    auto carve = [&](size_t bytes) -> char* {
        char* p = ws + off;
        off = (off + bytes + 255) & ~(size_t)255;
        return p;
    };
    float* hbuf = (float*)carve((size_t)N0 * 192 * sizeof(float)); // GEMM outputs
    float* abuf = (float*)carve((size_t)N1 * 192 * sizeof(float)); // aggregate outputs
    float* elb  = (float*)carve((size_t)N0 * 3 * sizeof(float));
    float* erb  = (float*)carve((size_t)N1 * 3 * sizeof(float));
    int*   offs = (int*)  carve((size_t)(N1 + 1) * sizeof(int));

    const int T = 256;

    // ---------------- Layer 0: 200000x256 @ 256x192, H=3, D=64 -------------
    launch_gemm<2, 4, false>(x, W0, hbuf, N0, 256, 192, stream);
    hipLaunchKernelGGL(attn_logits, dim3((N0 * 3 + T - 1) / T), dim3(T), 0, stream,
                       hbuf, al0, elb, N0, 3, 64);
    hipLaunchKernelGGL(attn_logits, dim3((N1 * 3 + T - 1) / T), dim3(T), 0, stream,
                       hbuf, ar0, erb, N1, 3, 64);
    hipLaunchKernelGGL(build_offsets, dim3((E0 + T - 1) / T), dim3(T), 0, stream,
                       dst0, E0, N1, offs);
    hipLaunchKernelGGL((gat_aggregate<3, 64>), dim3(N1), dim3(32), 0, stream,
                       hbuf, elb, erb, src0, offs, abuf, N1, 1);

    // ---------------- Layer 1: 100000x192 @ 192x192, H=3, D=64 -------------
    launch_gemm<2, 4, false>(abuf, W1, hbuf, N1, 192, 192, stream);
    hipLaunchKernelGGL(attn_logits, dim3((N1 * 3 + T - 1) / T), dim3(T), 0, stream,
                       hbuf, al1, elb, N1, 3, 64);
    hipLaunchKernelGGL(attn_logits, dim3((N2 * 3 + T - 1) / T), dim3(T), 0, stream,
                       hbuf, ar1, erb, N2, 3, 64);
    hipLaunchKernelGGL(build_offsets, dim3((E1 + T - 1) / T), dim3(T), 0, stream,
                       dst1, E1, N2, offs);
    hipLaunchKernelGGL((gat_aggregate<3, 64>), dim3(N2), dim3(32), 0, stream,
                       hbuf, elb, erb, src1, offs, abuf, N2, 1);

    // ---------------- Layer 2: 50000x192 @ 192x40, H=1, D=40 ---------------
    launch_gemm<1, 3, true>(abuf, W2, hbuf, N2, 192, 40, stream);
    hipLaunchKernelGGL(attn_logits, dim3((N2 + T - 1) / T), dim3(T), 0, stream,
                       hbuf, al2, elb, N2, 1, 40);
    hipLaunchKernelGGL(attn_logits, dim3((N3 + T - 1) / T), dim3(T), 0, stream,
                       hbuf, ar2, erb, N3, 1, 40);
    hipLaunchKernelGGL(build_offsets, dim3((E2 + T - 1) / T), dim3(T), 0, stream,
                       dst2, E2, N3, offs);
    hipLaunchKernelGGL((gat_aggregate<1, 40>), dim3(N3), dim3(32), 0, stream,
                       hbuf, elb, erb, src2, offs, outp, N3, 0);
}